// SetCriterion_11922829214444
// MI455X (gfx1250) — compile-verified
//
#include <hip/hip_runtime.h>
#include <math.h>

// ---------------------------------------------------------------------------
// DETR-style SetCriterion losses for MI455X (gfx1250, wave32).
//   Pass A: bandwidth-bound streaming focal-loss reduction over 41.9 MB logits
//           (float4 loads; ~1.8 us at 23.3 TB/s is the floor).
//   Pass B: 8192 matched pairs: focal correction + L1 bbox + angle + rotated IoU.
//   Cross-lane reduction uses V_WMMA_F32_16X16X4_F32 (the only full-f32 WMMA),
//   with B = ones so D row-sums collapse 2 f32 accumulators/lane in one XDL op.
// ---------------------------------------------------------------------------

typedef __attribute__((ext_vector_type(2))) float v2f;
typedef __attribute__((ext_vector_type(8))) float v8f;

#define Q_DIM 900
#define C_DIM 91
#define PI_F  3.14159265358979323846f

// Full-wave (32 lane) sum of (a0 + a1) per lane via f32 WMMA ones-reduction.
// A is 16x4 f32: lane l<16 holds (M=l,K=0/1), lane l>=16 holds (M=l-16,K=2/3).
// With B = ones(4x16): D[m][*] = rowsum(m) = a0[m]+a1[m]+a0[m+16]+a1[m+16].
// Lanes 0-15 see rows 0..7 in C[0..7]; lanes 16-31 see rows 8..15.
// Sum of C[0..7] + shfl_xor(16) -> full wave total in every lane.
__device__ __forceinline__ float wave_sum2(float a0, float a1) {
    v2f A;  A[0] = a0;   A[1] = a1;
    v2f Bm; Bm[0] = 1.0f; Bm[1] = 1.0f;
    v8f C = {};
    C = __builtin_amdgcn_wmma_f32_16x16x4_f32(
            /*neg_a=*/false, A, /*neg_b=*/false, Bm,
            /*c_mod=*/(short)0, C, /*reuse_a=*/false, /*reuse_b=*/false);
    float t = (C[0] + C[1]) + (C[2] + C[3]) + (C[4] + C[5]) + (C[6] + C[7]);
    t += __shfl_xor(t, 16);
    return t;
}

// Focal term with onehot = 0:  0.75 * sigmoid(x)^2 * softplus(x)
__device__ __forceinline__ float focal_neg(float x) {
    float ax  = fabsf(x);
    float e   = __expf(-ax);                 // e^{-|x|} in (0,1]
    float sp  = fmaxf(x, 0.0f) + __logf(1.0f + e);   // softplus(x)
    float inv = 1.0f / (1.0f + e);
    float sig = (x >= 0.0f) ? inv : e * inv;          // sigmoid(x)
    return 0.75f * sig * sig * sp;
}

// focal(onehot=1) - focal(onehot=0) at one logit (sparse correction).
// focal1 = 0.25 * (1-sigmoid)^2 * softplus(-x),  softplus(-x) = softplus(x) - x
__device__ __forceinline__ float focal_delta(float x) {
    float ax  = fabsf(x);
    float e   = __expf(-ax);
    float sp  = fmaxf(x, 0.0f) + __logf(1.0f + e);
    float inv = 1.0f / (1.0f + e);
    float sig = (x >= 0.0f) ? inv : e * inv;
    float f0  = 0.75f * sig * sig * sp;
    float om  = 1.0f - sig;
    float f1  = 0.25f * om * om * (sp - x);
    return f1 - f0;
}

// Rotated-rect IoU via Sutherland-Hodgman clipping (mirrors reference quirks:
// t=0 when |denom|<=1e-12, emit order inter-then-next, vertex cap 8).
__device__ float obb_iou(float cx1, float cy1, float w1, float h1, float th1,
                         float cx2, float cy2, float w2, float h2, float th2) {
    const float dxs[4] = {0.5f, -0.5f, -0.5f, 0.5f};
    const float dys[4] = {0.5f, 0.5f, -0.5f, -0.5f};
    float c1, s1, c2, s2;
    sincosf(th1, &s1, &c1);
    sincosf(th2, &s2, &c2);

    float px[8], py[8], qx[4], qy[4];
#pragma unroll
    for (int k = 0; k < 4; ++k) {
        float dx = dxs[k] * w1, dy = dys[k] * h1;
        px[k] = cx1 + c1 * dx - s1 * dy;
        py[k] = cy1 + s1 * dx + c1 * dy;
        dx = dxs[k] * w2; dy = dys[k] * h2;
        qx[k] = cx2 + c2 * dx - s2 * dy;
        qy[k] = cy2 + s2 * dx + c2 * dy;
    }
    float cqx = 0.25f * (qx[0] + qx[1] + qx[2] + qx[3]);
    float cqy = 0.25f * (qy[0] + qy[1] + qy[2] + qy[3]);

    int n = 4;
    for (int k = 0; k < 4; ++k) {
        if (n == 0) break;
        float ax = qx[k], ay = qy[k];
        int   k2 = (k + 1) & 3;
        float ex = qx[k2] - ax, ey = qy[k2] - ay;
        float sgn = ((ex * (cqy - ay) - ey * (cqx - ax)) >= 0.0f) ? 1.0f : -1.0f;

        float d[8];
        for (int i = 0; i < n; ++i)
            d[i] = sgn * (ex * (py[i] - ay) - ey * (px[i] - ax));

        float ox[8], oy[8];
        int m = 0;
        for (int i = 0; i < n; ++i) {
            int   j  = (i + 1 == n) ? 0 : i + 1;
            float dc = d[i], dn = d[j];
            bool  ci = dc >= 0.0f, ni = dn >= 0.0f;
            if (ci != ni && m < 8) {
                float den = dc - dn;
                float t   = (fabsf(den) > 1e-12f) ? dc / den : 0.0f;
                ox[m] = px[i] + t * (px[j] - px[i]);
                oy[m] = py[i] + t * (py[j] - py[i]);
                ++m;
            }
            if (ni && m < 8) { ox[m] = px[j]; oy[m] = py[j]; ++m; }
        }
        n = m;
        for (int i = 0; i < n; ++i) { px[i] = ox[i]; py[i] = oy[i]; }
    }

    float inter = 0.0f;
    for (int i = 0; i < n; ++i) {
        int j = (i + 1 == n) ? 0 : i + 1;
        inter += px[i] * py[j] - px[j] * py[i];
    }
    inter = 0.5f * fabsf(inter);
    float a1  = w1 * h1, a2 = w2 * h2;
    float uni = a1 + a2 - inter;
    return (uni > 0.0f) ? inter / uni : 0.0f;
}

// --------------------------- kernels ---------------------------------------

__global__ void init_ws_kernel(float* __restrict__ ws) {
    if (threadIdx.x < 4) ws[threadIdx.x] = 0.0f;
}

// Pass A: dense focal sum assuming onehot == 0 everywhere. Bandwidth-bound.
__global__ __launch_bounds__(256)
void focal_sum_kernel(const float* __restrict__ logits, unsigned long long n4,
                      unsigned long long total, float* __restrict__ ws) {
    unsigned long long tid    = (unsigned long long)blockIdx.x * blockDim.x + threadIdx.x;
    unsigned long long stride = (unsigned long long)gridDim.x * blockDim.x;
    const float4* p = (const float4*)logits;

    float a0 = 0.0f, a1 = 0.0f;
    for (unsigned long long i = tid; i < n4; i += stride) {
        float4 v = p[i];
        a0 += focal_neg(v.x) + focal_neg(v.y);
        a1 += focal_neg(v.z) + focal_neg(v.w);
    }
    // scalar tail (total not multiple of 4)
    unsigned long long rem_base = n4 * 4ull;
    if (rem_base + tid < total) a0 += focal_neg(logits[rem_base + tid]);

    float w = wave_sum2(a0, a1);   // v_wmma_f32_16x16x4_f32 ones-reduction

    __shared__ float part[8];
    int lane = threadIdx.x & 31, wid = threadIdx.x >> 5;
    if (lane == 0) part[wid] = w;
    __syncthreads();
    if (threadIdx.x == 0) {
        float s = 0.0f;
        int nw = (int)(blockDim.x >> 5);
        for (int k = 0; k < nw; ++k) s += part[k];
        atomicAdd(&ws[0], s);
    }
}

// Pass B: matched pairs — sparse focal correction + bbox L1 + angle + 1-IoU.
__global__ __launch_bounds__(256)
void matched_kernel(const float* __restrict__ logits,
                    const float* __restrict__ pred_boxes,
                    const float* __restrict__ tgt_boxes,
                    const int* __restrict__ batch_idx,
                    const int* __restrict__ src_idx,
                    const int* __restrict__ tgt_labels,
                    int N, float* __restrict__ ws) {
    int i = blockIdx.x * blockDim.x + threadIdx.x;
    float dce = 0.0f, dbb = 0.0f, dgi = 0.0f, dan = 0.0f;
    if (i < N) {
        int b = batch_idx[i], q = src_idx[i], l = tgt_labels[i];
        unsigned long long bq = (unsigned long long)b * Q_DIM + (unsigned long long)q;

        float x = logits[bq * C_DIM + (unsigned long long)l];
        dce = focal_delta(x);

        const float* sb = pred_boxes + bq * 5ull;
        const float* tb = tgt_boxes + (unsigned long long)i * 5ull;
        float s0 = sb[0], s1 = sb[1], s2 = sb[2], s3 = sb[3], s4 = sb[4];
        float t0 = tb[0], t1 = tb[1], t2 = tb[2], t3 = tb[3], t4 = tb[4];

        dbb = fabsf(s0 - t0) + fabsf(s1 - t1) + fabsf(s2 - t2) + fabsf(s3 - t3);
        dan = fminf(fabsf(s4 - t4), fabsf(s4 - (t4 + PI_F)));
        dgi = 1.0f - obb_iou(s0, s1, s2, s3, s4, t0, t1, t2, t3, t4);
    }
    // WMMA-based wave reductions (EXEC uniform here: all lanes reach this).
    float wce = wave_sum2(dce, 0.0f);
    float wbb = wave_sum2(dbb, 0.0f);
    float wgi = wave_sum2(dgi, 0.0f);
    float wan = wave_sum2(dan, 0.0f);
    if ((threadIdx.x & 31) == 0) {
        atomicAdd(&ws[0], wce);
        atomicAdd(&ws[1], wbb);
        atomicAdd(&ws[2], wgi);
        atomicAdd(&ws[3], wan);
    }
}

__global__ void finalize_kernel(const float* __restrict__ ws,
                                const int* __restrict__ nb_ptr,
                                float* __restrict__ out) {
    if (blockIdx.x == 0 && threadIdx.x == 0) {
        float nb = (float)nb_ptr[0];
        out[0] = ws[0] / nb;          // loss_ce  (mean(1).sum()/nb*Q == total/nb)
        out[1] = ws[1] / nb * 5.0f;   // loss_bbox (BBOX_COEF)
        out[2] = ws[2] / nb * 2.0f;   // loss_giou (GIOU_COEF)
        out[3] = ws[3] / nb;          // loss_angle (ANGLE_COEF)
    }
}

// --------------------------- launcher --------------------------------------

extern "C" void kernel_launch(void* const* d_in, const int* in_sizes, int n_in,
                              void* d_out, int out_size, void* d_ws, size_t ws_size,
                              hipStream_t stream) {
    const float* logits = (const float*)d_in[0];   // (128,900,91)
    const float* pboxes = (const float*)d_in[1];   // (128,900,5)
    const float* tboxes = (const float*)d_in[2];   // (N,5)
    const int*   bidx   = (const int*)d_in[3];     // (N,)
    const int*   sidx   = (const int*)d_in[4];     // (N,)
    const int*   tlab   = (const int*)d_in[5];     // (N,)
    const int*   nb     = (const int*)d_in[6];     // scalar
    float* ws  = (float*)d_ws;
    float* out = (float*)d_out;

    int N = in_sizes[3];
    unsigned long long total = (unsigned long long)in_sizes[0];
    unsigned long long n4    = total / 4ull;

    init_ws_kernel<<<1, 32, 0, stream>>>(ws);

    // 2048 blocks x 256 threads -> ~5 float4 iterations/lane over 10.48M elems.
    focal_sum_kernel<<<2048, 256, 0, stream>>>(logits, n4, total, ws);

    matched_kernel<<<(N + 255) / 256, 256, 0, stream>>>(
        logits, pboxes, tboxes, bidx, sidx, tlab, N, ws);

    finalize_kernel<<<1, 32, 0, stream>>>(ws, nb, out);
}